// MIFNet_40690520163033
// MI455X (gfx1250) — compile-verified
//
#include <hip/hip_runtime.h>
#include <hip/hip_bf16.h>
#include <math.h>

#define B_     32
#define T_     1000
#define CHANS_ 22
#define OUTC   64
#define DIN    128
#define DSTATE 16
#define POOLW  50
#define POOLS  15
#define NCLS   4
#define EPSF   1e-5f

typedef __attribute__((ext_vector_type(2))) float v2f;
typedef __attribute__((ext_vector_type(8))) float v8f;

// 16x16 output tile via V_WMMA_F32_16X16X4_F32, looping K in steps of 4.
// As: 16 x K (row-major, stride lda)   Bs: K x 16.. (row-major, stride ldb)
// Per ISA 7.12.2: A lane(lo,hi): a[j]=A[lo][2*hi+j]; B: b[j]=B[2*hi+j][lo];
// D: d[r]=D[8*hi+r][lo].
__device__ __forceinline__ v8f wmma_tile(const float* As, int lda,
                                         const float* Bs, int ldb,
                                         int K, int lane) {
  const int lo = lane & 15, hi = lane >> 4;
  v8f acc = {0.f,0.f,0.f,0.f,0.f,0.f,0.f,0.f};
  for (int k = 0; k < K; k += 4) {
    const int ka = k + 2*hi;
    v2f a, b;
    a.x = As[lo*lda + ka];
    a.y = As[lo*lda + ka + 1];
    b.x = Bs[ka*ldb + lo];
    b.y = Bs[(ka+1)*ldb + lo];
    acc = __builtin_amdgcn_wmma_f32_16x16x4_f32(false, a, false, b,
                                                (short)0, acc, false, false);
  }
  return acc;
}

__device__ __forceinline__ float siluf(float x){ return x / (1.f + __expf(-x)); }
__device__ __forceinline__ float softplusf(float x){ return (x > 20.f) ? x : log1pf(__expf(x)); }

// K1: grouped 1x1 conv (groups=2) + BN1 -> s1[b][c][t]
__global__ void k1_sconv_bn(const float* __restrict__ x, const float* __restrict__ w,
                            const float* __restrict__ g, const float* __restrict__ bta,
                            const float* __restrict__ m, const float* __restrict__ v,
                            float* __restrict__ s1) {
  int idx = blockIdx.x*256 + threadIdx.x;
  if (idx >= B_*DIN*T_) return;
  int t = idx % T_;
  int c = (idx / T_) % DIN;
  int b = idx / (T_*DIN);
  int grp = c >> 6;
  const float* xb = x + ((size_t)b*(2*CHANS_) + grp*CHANS_)*T_ + t;
  const float* wc = w + c*CHANS_;
  float acc = 0.f;
  #pragma unroll
  for (int i = 0; i < CHANS_; ++i) acc += xb[(size_t)i*T_]*wc[i];
  acc = (acc - m[c]) * rsqrtf(v[c]+EPSF) * g[c] + bta[c];
  s1[idx] = acc;
}

// K2: encoder GEMM (M=64,K=128) + bias + BN2 -> xin[n][64] (token-major)
__global__ void k2_enc(const float* __restrict__ s1, const float* __restrict__ encw,
                       const float* __restrict__ encb, const float* __restrict__ g,
                       const float* __restrict__ bta, const float* __restrict__ m,
                       const float* __restrict__ v, float* __restrict__ xin) {
  __shared__ float W2[OUTC*DIN];   // remapped enc weights, 32KB
  __shared__ float Bs[DIN*32];     // activation tile, 16KB
  const int tid = threadIdx.x;
  const int b  = blockIdx.x >> 5;
  const int t0 = (blockIdx.x & 31) * 32;
  for (int i = tid; i < OUTC*DIN; i += 256) {
    int o = i >> 7, c = i & 127;
    W2[i] = encw[((o*64 + (c & 63)) << 1) + (c >> 6)];
  }
  for (int i = tid; i < DIN*32; i += 256) {
    int c = i >> 5, n = i & 31;
    int t = t0 + n; if (t >= T_) t = T_-1;
    Bs[i] = s1[((size_t)b*DIN + c)*T_ + t];
  }
  __syncthreads();
  const int wave = tid >> 5, lane = tid & 31;
  const int mt = wave & 3, nt = wave >> 2;
  v8f d = wmma_tile(&W2[mt*16*DIN], DIN, &Bs[nt*16], 32, DIN, lane);
  const int lo = lane & 15, hi = lane >> 4;
  const int t = t0 + nt*16 + lo;
  if (t < T_) {
    size_t nb = ((size_t)b*T_ + t)*OUTC;
    #pragma unroll
    for (int r = 0; r < 8; ++r) {
      int mm = mt*16 + 8*hi + r;
      float val = d[r] + encb[mm];
      val = (val - m[mm]) * rsqrtf(v[mm]+EPSF) * g[mm] + bta[mm];
      xin[nb + mm] = val;
    }
  }
}

// K3: in_proj GEMM (M=256,K=64) -> xs[n][128], z[n][128]
__global__ void k3_inproj(const float* __restrict__ xin, const float* __restrict__ wproj,
                          float* __restrict__ xs, float* __restrict__ z) {
  __shared__ float Bs[64*16];
  const int tid = threadIdx.x;
  const int b = blockIdx.x / 63, t0 = (blockIdx.x % 63) * 16;
  for (int i = tid; i < 16*64; i += 256) {
    int n = i >> 6, k = i & 63;
    int t = t0 + n; if (t >= T_) t = T_-1;
    Bs[k*16 + n] = xin[((size_t)b*T_ + t)*OUTC + k];
  }
  __syncthreads();
  const int wave = tid >> 5, lane = tid & 31;
  const int lo = lane & 15, hi = lane >> 4;
  const int t = t0 + lo;
  for (int half = 0; half < 2; ++half) {
    int mt = wave*2 + half;  // 0..15
    v8f d = wmma_tile(wproj + (size_t)mt*16*64, 64, Bs, 16, 64, lane);
    if (t < T_) {
      size_t nb = ((size_t)b*T_ + t)*DIN;
      #pragma unroll
      for (int r = 0; r < 8; ++r) {
        int j = mt*16 + 8*hi + r;
        if (j < DIN) xs[nb + j] = d[r];
        else         z[nb + j - DIN] = d[r];
      }
    }
  }
}

// K4: depthwise conv3 (pad 1) + bias + SiLU -> u[n][128]
__global__ void k4_dwconv(const float* __restrict__ xs, const float* __restrict__ w,
                          const float* __restrict__ bias, float* __restrict__ u) {
  int idx = blockIdx.x*256 + threadIdx.x;
  if (idx >= B_*T_*DIN) return;
  int d = idx & 127;
  int n = idx >> 7;
  int t = n % T_;
  const float* wd = w + d*3;
  float acc = bias[d];
  if (t > 0)      acc += xs[(size_t)(n-1)*DIN + d] * wd[0];
  acc += xs[(size_t)n*DIN + d] * wd[1];
  if (t < T_-1)   acc += xs[(size_t)(n+1)*DIN + d] * wd[2];
  u[idx] = siluf(acc);
}

// K5: x_proj GEMM (M=36 padded to 48, K=128) -> dts[n][4], Bm[n][16], Cm[n][16]
__global__ void k5_xproj(const float* __restrict__ u, const float* __restrict__ wx,
                         float* __restrict__ dts, float* __restrict__ Bm,
                         float* __restrict__ Cm) {
  __shared__ float As[48*DIN];   // padded weights, 24KB
  __shared__ float Bs[DIN*16];   // 8KB
  const int tid = threadIdx.x;   // 128 threads
  const int b = blockIdx.x / 63, t0 = (blockIdx.x % 63) * 16;
  for (int i = tid; i < 48*DIN; i += 128) {
    int mrow = i >> 7, k = i & 127;
    As[i] = (mrow < 36) ? wx[mrow*DIN + k] : 0.f;
  }
  for (int i = tid; i < 16*DIN; i += 128) {
    int n = i >> 7, k = i & 127;
    int t = t0 + n; if (t >= T_) t = T_-1;
    Bs[k*16 + n] = u[((size_t)b*T_ + t)*DIN + k];
  }
  __syncthreads();
  const int wave = tid >> 5, lane = tid & 31;
  if (wave < 3) {
    v8f d = wmma_tile(&As[wave*16*DIN], DIN, Bs, 16, DIN, lane);
    const int lo = lane & 15, hi = lane >> 4;
    const int t = t0 + lo;
    if (t < T_) {
      size_t nb = (size_t)b*T_ + t;
      #pragma unroll
      for (int r = 0; r < 8; ++r) {
        int j = wave*16 + 8*hi + r;
        if (j < 4)       dts[nb*4 + j] = d[r];
        else if (j < 20) Bm[nb*16 + (j-4)] = d[r];
        else if (j < 36) Cm[nb*16 + (j-20)] = d[r];
      }
    }
  }
}

// K5b: dt_proj GEMM (M=128,K=4) + bias + softplus -> delta[n][128]
__global__ void k5b_dtproj(const float* __restrict__ dts, const float* __restrict__ wdt,
                           const float* __restrict__ bdt, float* __restrict__ delta) {
  __shared__ float As[DIN*4];
  __shared__ float Bs[4*16];
  const int tid = threadIdx.x;   // 256
  const int b = blockIdx.x / 63, t0 = (blockIdx.x % 63) * 16;
  for (int i = tid; i < DIN*4; i += 256) As[i] = wdt[i];
  if (tid < 64) {
    int n = tid >> 2, k = tid & 3;
    int t = t0 + n; if (t >= T_) t = T_-1;
    Bs[k*16 + n] = dts[((size_t)b*T_ + t)*4 + k];
  }
  __syncthreads();
  const int wave = tid >> 5, lane = tid & 31;
  v8f d = wmma_tile(&As[wave*16*4], 4, Bs, 16, 4, lane);
  const int lo = lane & 15, hi = lane >> 4;
  const int t = t0 + lo;
  if (t < T_) {
    size_t nb = ((size_t)b*T_ + t)*DIN;
    #pragma unroll
    for (int r = 0; r < 8; ++r) {
      int j = wave*16 + 8*hi + r;
      delta[nb + j] = softplusf(d[r] + bdt[j]);
    }
  }
}

// K6: selective scan + LayerNorm + SiLU gate. One block per batch, 128 threads.
__global__ void k6_scan(const float* __restrict__ u, const float* __restrict__ delta,
                        const float* __restrict__ Bm, const float* __restrict__ Cm,
                        const float* __restrict__ A_log, const float* __restrict__ Dp,
                        const float* __restrict__ lng, const float* __restrict__ lnb,
                        const float* __restrict__ z, float* __restrict__ yact) {
  __shared__ float sB[DSTATE], sC[DSTATE];
  __shared__ float red[8];
  const int d = threadIdx.x;     // 0..127 = channel
  const int b = blockIdx.x;
  const int lane = d & 31, wave = d >> 5;
  float Arow[DSTATE], h[DSTATE];
  #pragma unroll
  for (int n2 = 0; n2 < DSTATE; ++n2) {
    Arow[n2] = -__expf(A_log[d*DSTATE + n2]);
    h[n2] = 0.f;
  }
  const float Dd = Dp[d], gg = lng[d], bb = lnb[d];
  for (int t = 0; t < T_; ++t) {
    size_t nb = (size_t)b*T_ + t;
    if (d < DSTATE)        sB[d]        = Bm[nb*DSTATE + d];
    else if (d < 2*DSTATE) sC[d-DSTATE] = Cm[nb*DSTATE + (d-DSTATE)];
    __syncthreads();
    const float dl = delta[nb*DIN + d];
    const float uu = u[nb*DIN + d];
    float acc = 0.f;
    #pragma unroll
    for (int n2 = 0; n2 < DSTATE; ++n2) {
      float dA = __expf(dl * Arow[n2]);
      h[n2] = dA*h[n2] + dl*sB[n2]*uu;
      acc += h[n2]*sC[n2];
    }
    const float y = acc + uu*Dd;
    // LayerNorm over the 128 channels: wave32 shuffle tree + 4 wave partials
    float s = y, s2 = y*y;
    #pragma unroll
    for (int off = 16; off > 0; off >>= 1) {
      s  += __shfl_xor(s,  off, 32);
      s2 += __shfl_xor(s2, off, 32);
    }
    if (lane == 0) { red[wave] = s; red[4+wave] = s2; }
    __syncthreads();
    const float ts  = red[0]+red[1]+red[2]+red[3];
    const float ts2 = red[4]+red[5]+red[6]+red[7];
    const float mu  = ts * (1.f/DIN);
    const float var = ts2*(1.f/DIN) - mu*mu;
    const float rstd = rsqrtf(var + EPSF);
    const float yn = (y - mu)*rstd*gg + bb;
    yact[nb*DIN + d] = yn * siluf(z[nb*DIN + d]);
    __syncthreads();   // protect sB/sC/red before next iteration
  }
}

// K7: out_proj GEMM (M=64,K=128) -> yo[n][64]
__global__ void k7_outproj(const float* __restrict__ yact, const float* __restrict__ wout,
                           float* __restrict__ yo) {
  __shared__ float Bs[DIN*32];   // 16KB
  const int tid = threadIdx.x;
  const int b = blockIdx.x >> 5, t0 = (blockIdx.x & 31) * 32;
  for (int i = tid; i < 32*DIN; i += 256) {
    int n = i >> 7, k = i & 127;
    int t = t0 + n; if (t >= T_) t = T_-1;
    Bs[k*32 + n] = yact[((size_t)b*T_ + t)*DIN + k];
  }
  __syncthreads();
  const int wave = tid >> 5, lane = tid & 31;
  const int mt = wave & 3, nt = wave >> 2;
  v8f d = wmma_tile(wout + (size_t)mt*16*DIN, DIN, &Bs[nt*16], 32, DIN, lane);
  const int lo = lane & 15, hi = lane >> 4;
  const int t = t0 + nt*16 + lo;
  if (t < T_) {
    size_t nb = ((size_t)b*T_ + t)*OUTC;
    #pragma unroll
    for (int r = 0; r < 8; ++r) yo[nb + mt*16 + 8*hi + r] = d[r];
  }
}

// K8: average pool window 50 stride 15 -> yp[b][o][p], p<64
__global__ void k8_pool(const float* __restrict__ yo, float* __restrict__ yp) {
  int idx = blockIdx.x*256 + threadIdx.x;
  if (idx >= B_*64*64) return;
  int o = idx & 63;
  int p = (idx >> 6) & 63;
  int b = idx >> 12;
  const float* src = yo + ((size_t)b*T_ + p*POOLS)*OUTC + o;
  float acc = 0.f;
  #pragma unroll 10
  for (int w2 = 0; w2 < POOLW; ++w2) acc += src[(size_t)w2*OUTC];
  yp[((size_t)b*64 + o)*64 + p] = acc * (1.f/POOLW);
}

// K9: norm-clipped classifier -> out[b][4]
__global__ void k9_cls(const float* __restrict__ yp, const float* __restrict__ cw,
                       const float* __restrict__ cb, float* __restrict__ out) {
  __shared__ float red[256];
  __shared__ float scale[NCLS];
  const int tid = threadIdx.x;
  for (int c = 0; c < NCLS; ++c) {
    float p = 0.f;
    for (int j = tid; j < 4096; j += 256) { float w = cw[c*4096 + j]; p += w*w; }
    red[tid] = p; __syncthreads();
    for (int off = 128; off > 0; off >>= 1) {
      if (tid < off) red[tid] += red[tid+off];
      __syncthreads();
    }
    if (tid == 0) { float rn = sqrtf(red[0]); scale[c] = fminf(1.f, 0.5f/(rn + 1e-7f)); }
    __syncthreads();
  }
  if (tid < B_*NCLS) {
    int c = tid & 3, b = tid >> 2;
    const float* yb = yp + (size_t)b*4096;
    const float* wc = cw + (size_t)c*4096;
    float acc = 0.f;
    for (int j = 0; j < 4096; ++j) acc += yb[j]*wc[j];
    out[b*NCLS + c] = acc*scale[c] + cb[c];
  }
}

extern "C" void kernel_launch(void* const* d_in, const int* in_sizes, int n_in,
                              void* d_out, int out_size, void* d_ws, size_t ws_size,
                              hipStream_t stream) {
  const float* x        = (const float*)d_in[0];
  const float* sconv_w  = (const float*)d_in[1];
  const float* bn1_g    = (const float*)d_in[2];
  const float* bn1_b    = (const float*)d_in[3];
  const float* bn1_m    = (const float*)d_in[4];
  const float* bn1_v    = (const float*)d_in[5];
  const float* enc_w    = (const float*)d_in[6];
  const float* enc_b    = (const float*)d_in[7];
  const float* bn2_g    = (const float*)d_in[8];
  const float* bn2_b    = (const float*)d_in[9];
  const float* bn2_m    = (const float*)d_in[10];
  const float* bn2_v    = (const float*)d_in[11];
  const float* in_proj  = (const float*)d_in[12];
  const float* conv1d_w = (const float*)d_in[13];
  const float* conv1d_b = (const float*)d_in[14];
  const float* x_proj_w = (const float*)d_in[15];
  const float* dt_proj_w= (const float*)d_in[16];
  const float* dt_proj_b= (const float*)d_in[17];
  const float* A_log    = (const float*)d_in[18];
  const float* Dp       = (const float*)d_in[19];
  const float* ln_g     = (const float*)d_in[20];
  const float* ln_b     = (const float*)d_in[21];
  const float* out_proj = (const float*)d_in[22];
  const float* cls_w    = (const float*)d_in[23];
  const float* cls_b    = (const float*)d_in[24];
  float* out = (float*)d_out;

  float* ws = (float*)d_ws;
  size_t o = 0;
  float* s1    = ws + o; o += (size_t)B_*DIN*T_;     // (b,c,t)
  float* xin   = ws + o; o += (size_t)B_*T_*OUTC;    // token-major
  float* xs    = ws + o; o += (size_t)B_*T_*DIN;
  float* zg    = ws + o; o += (size_t)B_*T_*DIN;
  float* u     = ws + o; o += (size_t)B_*T_*DIN;
  float* dts   = ws + o; o += (size_t)B_*T_*4;
  float* Bmw   = ws + o; o += (size_t)B_*T_*DSTATE;
  float* Cmw   = ws + o; o += (size_t)B_*T_*DSTATE;
  float* delta = ws + o; o += (size_t)B_*T_*DIN;
  float* yact  = ws + o; o += (size_t)B_*T_*DIN;
  float* yo    = ws + o; o += (size_t)B_*T_*OUTC;
  float* yp    = ws + o; o += (size_t)B_*64*64;

  k1_sconv_bn<<<(B_*DIN*T_ + 255)/256, 256, 0, stream>>>(x, sconv_w, bn1_g, bn1_b, bn1_m, bn1_v, s1);
  k2_enc     <<<B_*32,  256, 0, stream>>>(s1, enc_w, enc_b, bn2_g, bn2_b, bn2_m, bn2_v, xin);
  k3_inproj  <<<B_*63,  256, 0, stream>>>(xin, in_proj, xs, zg);
  k4_dwconv  <<<(B_*T_*DIN + 255)/256, 256, 0, stream>>>(xs, conv1d_w, conv1d_b, u);
  k5_xproj   <<<B_*63,  128, 0, stream>>>(u, x_proj_w, dts, Bmw, Cmw);
  k5b_dtproj <<<B_*63,  256, 0, stream>>>(dts, dt_proj_w, dt_proj_b, delta);
  k6_scan    <<<B_,     128, 0, stream>>>(u, delta, Bmw, Cmw, A_log, Dp, ln_g, ln_b, zg, yact);
  k7_outproj <<<B_*32,  256, 0, stream>>>(yact, out_proj, yo);
  k8_pool    <<<(B_*64*64 + 255)/256, 256, 0, stream>>>(yo, yp);
  k9_cls     <<<1,      256, 0, stream>>>(yp, cls_w, cls_b, out);
}